// PerAgentMLP_53137335386139
// MI455X (gfx1250) — compile-verified
//
#include <hip/hip_runtime.h>
#include <cstdint>

#define NAGENTS_MAX 128
#define OBS 624
#define HID 256
#define OUTD 24
#define LIMBS 12
#define LIMB_SZ 52
#define XS 644   // padded LDS row stride for X / H2 region (bank = (4r+k)%64, conflict-free)
#define HS 260   // padded LDS row stride for H (same property)

typedef __attribute__((ext_vector_type(2))) float v2f;
typedef __attribute__((ext_vector_type(8))) float v8f;

// ---------------- grouping kernels (build per-agent sample lists) ----------------

__global__ void zero_ws_kernel(int* counts, int* cursor) {
  int t = threadIdx.x;
  counts[t] = 0;
  cursor[t] = 0;
}

__global__ void hist_kernel(const int* __restrict__ ids, int* counts, int n) {
  int i = blockIdx.x * blockDim.x + threadIdx.x;
  if (i < n) atomicAdd(&counts[ids[i]], 1);
}

__global__ void scan_kernel(const int* __restrict__ counts, int* offsets, int na) {
  if (threadIdx.x == 0) {
    int s = 0;
    for (int a = 0; a < na; ++a) { offsets[a] = s; s += counts[a]; }
    offsets[na] = s;
  }
}

__global__ void scatter_kernel(const int* __restrict__ ids, const int* __restrict__ offsets,
                               int* cursor, int* order, int n) {
  int i = blockIdx.x * blockDim.x + threadIdx.x;
  if (i < n) {
    int id = ids[i];
    int p = atomicAdd(&cursor[id], 1);
    order[offsets[id] + p] = i;
  }
}

// ---------------- main fused MLP kernel ----------------
// grid = (N_AGENTS, maxRowTiles); block = 128 (4 x wave32)
// One workgroup handles a 16-sample row tile of one agent through all 3 layers.

__global__ __launch_bounds__(128)
void mlp_kernel(const float* __restrict__ obs,
                const unsigned char* __restrict__ obs_mask,   // jnp.bool_ -> 1 byte
                const float* __restrict__ Wi_all, const float* __restrict__ Bi_all,
                const float* __restrict__ Wh_all, const float* __restrict__ Bh_all,
                const float* __restrict__ Wo_all, const float* __restrict__ Bo_all,
                const int* __restrict__ offsets, const int* __restrict__ order,
                float* __restrict__ out)
{
  __shared__ float sX[16 * XS];   // X for layer1; aliased as H2 (stride HS) for layer2 output
  __shared__ float sH[16 * HS];
  __shared__ float sInv[16];
  __shared__ int   sRow[16];

  const int agent = blockIdx.x;
  const int start = offsets[agent];
  const int end   = offsets[agent + 1];
  const int t0    = start + (int)blockIdx.y * 16;
  if (t0 >= end) return;                       // uniform per block
  const int nrows = min(16, end - t0);

  const int tid   = threadIdx.x;
  const int lane  = tid & 31;
  const int wv    = tid >> 5;                  // wave id 0..3
  const int m     = lane & 15;                 // A-row / B-col within 16
  const int khalf = lane >> 4;                 // 0: K=0,1 ; 1: K=2,3
  const int k0    = khalf * 2;

  const float* Wi = Wi_all + (size_t)agent * OBS * HID;
  const float* Bi = Bi_all + (size_t)agent * HID;
  const float* Wh = Wh_all + (size_t)agent * HID * HID;
  const float* Bh = Bh_all + (size_t)agent * HID;
  const float* Wo = Wo_all + (size_t)agent * HID * OUTD;
  const float* Bo = Bo_all + (size_t)agent * OUTD;

  // row indices + 1/keep_sum per row
  if (tid < 16) {
    int s = (tid < nrows) ? order[t0 + tid] : -1;
    sRow[tid] = s;
    float inv = 1.0f;
    if (s >= 0) {
      float sum = 0.f;
      #pragma unroll
      for (int l = 0; l < LIMBS; ++l)
        sum += obs_mask[s * LIMBS + l] ? 0.f : 1.f;
      inv = 1.0f / sum;   // matches reference (inf if all limbs masked)
    }
    sInv[tid] = inv;
  }
  __syncthreads();

  // stage masked X tile [16 x 624] into LDS
  for (int idx = tid; idx < 16 * OBS; idx += 128) {
    int r = idx / OBS;
    int c = idx - r * OBS;
    int s = sRow[r];
    float v = 0.f;
    if (s >= 0) {
      v = obs[(size_t)s * OBS + c];
      if (obs_mask[s * LIMBS + c / LIMB_SZ]) v = 0.f;
    }
    sX[r * XS + c] = v;
  }
  __syncthreads();

  // ---------------- layer 1: H = relu(X @ Wi * inv + Bi), K = 624 ----------------
  {
    const int cb = wv * 64;                    // this wave's 64-column slab
    v8f acc[4] = {};
    for (int kk = 0; kk < OBS; kk += 4) {
      v2f a;
      a.x = sX[m * XS + kk + k0];
      a.y = sX[m * XS + kk + k0 + 1];
      const float* w = Wi + (size_t)(kk + k0) * HID + cb + m;
      #pragma unroll
      for (int t = 0; t < 4; ++t) {
        v2f b;
        b.x = w[t * 16];
        b.y = w[t * 16 + HID];
        acc[t] = __builtin_amdgcn_wmma_f32_16x16x4_f32(
            false, a, false, b, (short)0, acc[t], false, false);
      }
    }
    #pragma unroll
    for (int t = 0; t < 4; ++t) {
      int col = cb + t * 16 + m;
      float bias = Bi[col];
      #pragma unroll
      for (int j = 0; j < 8; ++j) {
        int r = j + 8 * khalf;
        float h = acc[t][j] * sInv[r] + bias;
        sH[r * HS + col] = h > 0.f ? h : 0.f;
      }
    }
  }
  __syncthreads();

  float* sH2 = sX;   // reuse X region for H2 (stride HS)

  // ---------------- layer 2: H2 = relu(H @ Wh + Bh), K = 256 ----------------
  {
    const int cb = wv * 64;
    v8f acc[4] = {};
    for (int kk = 0; kk < HID; kk += 4) {
      v2f a;
      a.x = sH[m * HS + kk + k0];
      a.y = sH[m * HS + kk + k0 + 1];
      const float* w = Wh + (size_t)(kk + k0) * HID + cb + m;
      #pragma unroll
      for (int t = 0; t < 4; ++t) {
        v2f b;
        b.x = w[t * 16];
        b.y = w[t * 16 + HID];
        acc[t] = __builtin_amdgcn_wmma_f32_16x16x4_f32(
            false, a, false, b, (short)0, acc[t], false, false);
      }
    }
    #pragma unroll
    for (int t = 0; t < 4; ++t) {
      int col = cb + t * 16 + m;
      float bias = Bh[col];
      #pragma unroll
      for (int j = 0; j < 8; ++j) {
        int r = j + 8 * khalf;
        float h = acc[t][j] + bias;
        sH2[r * HS + col] = h > 0.f ? h : 0.f;
      }
    }
  }
  __syncthreads();

  // ---------------- layer 3: out = H2 @ Wo + Bo, K = 256, N = 24 ----------------
  if (wv < 2) {                                // wave-uniform: EXEC all-ones inside
    int col = wv * 16 + m;
    bool cok = col < OUTD;
    v8f acc = {};
    for (int kk = 0; kk < HID; kk += 4) {
      v2f a;
      a.x = sH2[m * HS + kk + k0];
      a.y = sH2[m * HS + kk + k0 + 1];
      v2f b;
      b.x = cok ? Wo[(size_t)(kk + k0) * OUTD + col] : 0.f;
      b.y = cok ? Wo[(size_t)(kk + k0 + 1) * OUTD + col] : 0.f;
      acc = __builtin_amdgcn_wmma_f32_16x16x4_f32(
          false, a, false, b, (short)0, acc, false, false);
    }
    float bo = cok ? Bo[col] : 0.f;
    #pragma unroll
    for (int j = 0; j < 8; ++j) {
      int r = j + 8 * khalf;
      if (cok && r < nrows)
        out[(size_t)sRow[r] * OUTD + col] = acc[j] + bo;
    }
  }
}

// ---------------- launch ----------------

extern "C" void kernel_launch(void* const* d_in, const int* in_sizes, int n_in,
                              void* d_out, int out_size, void* d_ws, size_t ws_size,
                              hipStream_t stream) {
  const float*         obs      = (const float*)d_in[0];
  const unsigned char* obs_mask = (const unsigned char*)d_in[1];
  const int*           ids      = (const int*)d_in[2];
  const float*         Wi       = (const float*)d_in[3];
  const float*         Bi       = (const float*)d_in[4];
  const float*         Wh       = (const float*)d_in[5];
  const float*         Bh       = (const float*)d_in[6];
  const float*         Wo       = (const float*)d_in[7];
  const float*         Bo       = (const float*)d_in[8];
  float*               out      = (float*)d_out;

  const int Bn = in_sizes[2];            // batch size (count of unimal_ids)
  const int na = in_sizes[4] / HID;      // number of agents

  int* counts  = (int*)d_ws;
  int* cursor  = counts + NAGENTS_MAX;
  int* offsets = cursor + NAGENTS_MAX;
  int* order   = offsets + (NAGENTS_MAX + 1);

  zero_ws_kernel<<<1, NAGENTS_MAX, 0, stream>>>(counts, cursor);
  hist_kernel<<<(Bn + 255) / 256, 256, 0, stream>>>(ids, counts, Bn);
  scan_kernel<<<1, 32, 0, stream>>>(counts, offsets, na);
  scatter_kernel<<<(Bn + 255) / 256, 256, 0, stream>>>(ids, offsets, cursor, order, Bn);

  const int maxTiles = (Bn + 15) / 16;
  dim3 grid(na, maxTiles);
  mlp_kernel<<<grid, 128, 0, stream>>>(obs, obs_mask, Wi, Bi, Wh, Bh, Wo, Bo,
                                       offsets, order, out);
}